// Mixture_33749853012533
// MI455X (gfx1250) — compile-verified
//
#include <hip/hip_runtime.h>
#include <hip/hip_bf16.h>

// ---------------------------------------------------------------------------
// Problem: B=64, T=128, D=64, K=64 online Bayesian mixture scan.
//   Kernel 1: proj[b,t,k] = sum_d data[b,t,d] * W[d,k]   (the only true GEMM)
//             -> V_WMMA_F32_16X16X4_F32 tiles.
//   Kernel 2: per-batch Sherman-Morrison scan over T (exact, no inversions):
//             A <- A - c u u^T,  A_bs <- A_bs + u (c(y - A_bs^T x))^T,
//             logdet <- logdet - log1p(x^T A x).
// ---------------------------------------------------------------------------

typedef float v2f __attribute__((ext_vector_type(2)));
typedef float v8f __attribute__((ext_vector_type(8)));

#define NB 64
#define NT 128
#define ND 64
#define NK 64

// ---------------------------------------------------------------------------
// Kernel 1: (B*T x D) * (D x K) GEMM via f32 WMMA 16x16x4.
// One 16x16 output tile per wave; 16 chained WMMAs over K(=D)=64.
// A frag (16x4): lanes 0-15 hold M=lane, K=kk+{0,1}; lanes 16-31 M=lane-16,
// K=kk+{2,3}. B frag (4x16): lanes 0-15 hold N=lane rows kk+{0,1}; lanes
// 16-31 hold N=lane-16 rows kk+{2,3}. C/D: VGPR r = row r (+8 for hi lanes).
// ---------------------------------------------------------------------------
__global__ __launch_bounds__(128) void proj_gemm_wmma(
    const float* __restrict__ data,   // (B*T, D) = (8192, 64)
    const float* __restrict__ W,      // (D, K)   = (64, 64)
    float* __restrict__ proj)         // (B*T, K) = (8192, 64)
{
  const int lane  = threadIdx.x & 31;
  const int wave  = threadIdx.x >> 5;
  const int tile  = blockIdx.x * 4 + wave;  // 0..2047
  const int mtile = tile >> 2;              // 512 row tiles over B*T
  const int ntile = tile & 3;               // 4 col tiles over K

  const int half = (lane < 16) ? 0 : 2;     // K sub-offset held by this half
  const int l15  = lane & 15;

  const float* arow = data + (mtile * 16 + l15) * ND;  // A-frag row (M = l15)
  const int    bcol = ntile * 16 + l15;                // B-frag col (N = l15)

  v8f acc = {};
#pragma unroll
  for (int kk = 0; kk < ND; kk += 4) {
    v2f a, bf;
    a.x  = arow[kk + half + 0];
    a.y  = arow[kk + half + 1];
    bf.x = W[(kk + half + 0) * NK + bcol];
    bf.y = W[(kk + half + 1) * NK + bcol];
    acc = __builtin_amdgcn_wmma_f32_16x16x4_f32(
        /*neg_a=*/false, a, /*neg_b=*/false, bf,
        /*c_mod=*/(short)0, acc, /*reuse_a=*/false, /*reuse_b=*/false);
  }

  const int mbase = mtile * 16 + ((lane < 16) ? 0 : 8);
#pragma unroll
  for (int r = 0; r < 8; ++r)
    proj[(mbase + r) * NK + (ntile * 16 + l15)] = acc[r];
}

// ---------------------------------------------------------------------------
// Kernel 2: one workgroup (256 threads = 8 waves) per batch element.
// All state LDS-resident (~55 KB of the 320 KB WGP LDS).
// ---------------------------------------------------------------------------
__global__ __launch_bounds__(256) void mixture_scan(
    const float* __restrict__ data,     // (B,T,D)
    const float* __restrict__ targets,  // (B,T)
    const float* __restrict__ Wg,       // (D,K)
    const float* __restrict__ proj,     // (B,T,K) from kernel 1
    float* __restrict__ preds)          // (B,T)
{
  constexpr float noise_var = 0.0625f;               // 0.25^2
  constexpr float minor_var = 0.0625f;               // (0.25/1)^2
  constexpr float log_half  = -0.69314718055994531f; // log(P0) = log1p(-P0)
  constexpr float inv_nv    = 1.0f / noise_var;
  constexpr float inv_mv    = 1.0f / minor_var;

  __shared__ float A[ND][ND + 1];     // (S_xx + I)^-1
  __shared__ float Abs[ND][NK + 1];   // A @ bs (posterior means)
  __shared__ float Wsh[ND][NK + 1];
  __shared__ float Sxy[ND], sse[NK], wns[NK];
  __shared__ float xs[ND], pr[NK], u[ND], coef[NK];
  __shared__ float red[4][64];
  __shared__ float red2[128];
  __shared__ float logw[128], wts[128];
  __shared__ float scal[4];           // 0:y  1:y2  2:logdet  3:c

  const int b      = blockIdx.x;
  const int tid    = threadIdx.x;
  const int part   = tid >> 6;   // 0..3
  const int lane64 = tid & 63;

  // ---- init state: S_xx=0 -> A=I, bs=W, A_bs=W, logdet=0 ----
  for (int idx = tid; idx < ND * NK; idx += 256) {
    const int i = idx >> 6, j = idx & 63;
    const float w = Wg[idx];
    A[i][j]   = (i == j) ? 1.0f : 0.0f;
    Wsh[i][j] = w;
    Abs[i][j] = w;
  }
  if (tid < 64) { Sxy[tid] = 0.0f; sse[tid] = 0.0f; }
  if (tid == 0) { scal[1] = 0.0f; scal[2] = 0.0f; }
  __syncthreads();
  if (tid < 64) {
    float s = 0.0f;
#pragma unroll 8
    for (int d = 0; d < ND; ++d) { const float w = Wsh[d][tid]; s += w * w; }
    wns[tid] = s;
  }
  __syncthreads();

  for (int t = 0; t < NT; ++t) {
    const long base = (long)(b * NT + t);
    if (tid < 64) {
      xs[tid] = data[base * ND + tid];
      pr[tid] = proj[base * NK + tid];
    }
    if (tid == 64) scal[0] = targets[base];
    if (tid == 65 && t + 1 < NT)
      __builtin_prefetch(&data[(base + 1) * ND], 0, 0);  // global_prefetch_b8
    __syncthreads();
    const float y = scal[0], y2 = scal[1], logdet = scal[2];

    // ---- quad_k = sum_i (Sxy[i] + W[i][k]) * Abs[i][k] ----
    {
      float qp = 0.0f;
      const int k = lane64;
#pragma unroll
      for (int ii = 0; ii < 16; ++ii) {
        const int i = part * 16 + ii;
        qp += (Sxy[i] + Wsh[i][k]) * Abs[i][k];
      }
      red[part][k] = qp;
    }
    __syncthreads();
    if (tid < 64) {
      const int k = tid;
      const float quad = red[0][k] + red[1][k] + red[2][k] + red[3][k];
      logw[64 + k] = -0.5f * (y2 * inv_nv + wns[k] * inv_mv - quad * inv_nv)
                     + log_half + 0.5f * logdet;         // D*log(TAU)=0
      logw[k] = -0.5f * sse[k] * inv_nv + log_half;
    }
    __syncthreads();

    // ---- softmax over 2K = 128 ----
    if (tid < 128) red2[tid] = logw[tid];
    __syncthreads();
    for (int s = 64; s > 0; s >>= 1) {
      if (tid < s) red2[tid] = fmaxf(red2[tid], red2[tid + s]);
      __syncthreads();
    }
    const float mx = red2[0];
    __syncthreads();
    if (tid < 128) {
      const float e = __expf(logw[tid] - mx);
      wts[tid]  = e;
      red2[tid] = e;
    }
    __syncthreads();
    for (int s = 64; s > 0; s >>= 1) {
      if (tid < s) red2[tid] += red2[tid + s];
      __syncthreads();
    }
    const float inv_sum = 1.0f / red2[0];
    __syncthreads();

    // ---- ws_d = sum_k locs[d][k] * wts[k]; pred = x . ws ----
    {
      const int d = lane64;
      float wp = 0.0f;
#pragma unroll
      for (int kk = 0; kk < 32; ++kk) {
        const int k = part * 32 + kk;
        const float loc = (k < 64) ? Wsh[d][k] : Abs[d][k - 64];
        wp += loc * wts[k];
      }
      red[part][d] = wp;
    }
    __syncthreads();
    if (tid < 64) {
      const float wsd = (red[0][tid] + red[1][tid] + red[2][tid] + red[3][tid]) * inv_sum;
      red2[tid] = xs[tid] * wsd;
    }
    __syncthreads();
    for (int s = 32; s > 0; s >>= 1) {
      if (tid < s) red2[tid] += red2[tid + s];
      __syncthreads();
    }
    if (tid == 0) preds[base] = red2[0];

    // ---- state update: u = A x ----
    {
      const int i = lane64;
      float up = 0.0f;
#pragma unroll
      for (int jj = 0; jj < 16; ++jj) {
        const int j = part * 16 + jj;
        up += A[i][j] * xs[j];
      }
      red[part][i] = up;
    }
    __syncthreads();
    if (tid < 64) {
      const float ui = red[0][tid] + red[1][tid] + red[2][tid] + red[3][tid];
      u[tid]    = ui;
      red2[tid] = ui * xs[tid];
    }
    __syncthreads();
    for (int s = 32; s > 0; s >>= 1) {
      if (tid < s) red2[tid] += red2[tid + s];
      __syncthreads();
    }
    if (tid == 0) {
      const float sxax = red2[0];
      scal[3] = 1.0f / (1.0f + sxax);
      scal[2] = logdet - log1pf(sxax);   // logdet(A) recursion
      scal[1] = y2 + y * y;              // exclusive -> include after output
    }
    __syncthreads();
    const float c = scal[3];

    // ---- v_k = x^T Abs[:,k]; coef_k = c*(y - v_k); sse/Sxy updates ----
    {
      const int k = lane64;
      float vp = 0.0f;
#pragma unroll
      for (int ii = 0; ii < 16; ++ii) {
        const int i = part * 16 + ii;
        vp += xs[i] * Abs[i][k];
      }
      red[part][k] = vp;
    }
    __syncthreads();
    if (tid < 64) {
      const float vk = red[0][tid] + red[1][tid] + red[2][tid] + red[3][tid];
      coef[tid] = c * (y - vk);
      const float r = y - pr[tid];
      sse[tid] += r * r;
      Sxy[tid] += xs[tid] * y;
    }
    __syncthreads();

    // ---- rank-1 updates: A -= c u u^T;  Abs += u coef^T ----
    for (int idx = tid; idx < ND * NK; idx += 256) {
      const int i = idx >> 6, j = idx & 63;
      const float ui = u[i];
      A[i][j]   -= c * ui * u[j];
      Abs[i][j] += ui * coef[j];
    }
    __syncthreads();
  }
}

extern "C" void kernel_launch(void* const* d_in, const int* in_sizes, int n_in,
                              void* d_out, int out_size, void* d_ws, size_t ws_size,
                              hipStream_t stream) {
  const float* data    = (const float*)d_in[0];  // (64,128,64) f32
  const float* targets = (const float*)d_in[1];  // (64,128)    f32
  const float* W       = (const float*)d_in[2];  // (64,64)     f32
  float* preds = (float*)d_out;                  // (64,128)    f32
  float* proj  = (float*)d_ws;                   // (8192,64)   f32 = 2 MB scratch

  // Kernel 1: 2048 16x16 tiles, 4 waves/block -> 512 blocks.
  proj_gemm_wmma<<<dim3(512), dim3(128), 0, stream>>>(data, W, proj);
  // Kernel 2: one workgroup per batch chain.
  mixture_scan<<<dim3(NB), dim3(256), 0, stream>>>(data, targets, W, proj, preds);
}